// CPModule_84275848282425
// MI455X (gfx1250) — compile-verified
//
#include <hip/hip_runtime.h>
#include <stdint.h>

// CP-decomposition (TensoRF) feature eval for MI455X / gfx1250.
// out[c][n] = lerp(p0[c], x_n) * lerp(p1[c], y_n) * lerp(p2[c], z_n)
//
// HBM-bound: 384MB output stream @ 23.3 TB/s is the floor (~17.5us).
// CDNA5 path: async global->LDS staging of the 172.8KB of tables
// (global_load_async_to_lds_b128 / s_wait_asynccnt), LDS transpose to a
// padded [300][52] layout for ds_load_b128 4-component gathers, 4 points
// per thread for b128 stores, non-temporal hints on the big streams.

#define NPTS      2000000
#define NCOMP     48
#define GRD       300
#define TRSTRIDE  52                       // padded floats per grid row (16B-aligned rows)
#define STAGE_F   (NCOMP * GRD)            // 14400 floats raw staging
#define TR_F      (GRD * TRSTRIDE)         // 15600 floats per transposed table
#define LDS_F     (STAGE_F + 3 * TR_F)     // 61200 floats = 244800 B  (<= 320KB WGP LDS)
#define THREADS   512

typedef __attribute__((ext_vector_type(4))) float f4;

__device__ __forceinline__ void axis_prep(float coord, int& o0, int& o1, float& w) {
    float pos = fmaf(coord, 149.5f, 149.5f);      // (coord+1)*0.5*(G-1)
    int i0 = (int)pos;                            // pos >= 0, trunc == floor
    i0 = i0 < 0 ? 0 : i0;
    i0 = i0 > (GRD - 1) ? (GRD - 1) : i0;
    int i1 = i0 + 1;
    i1 = i1 > (GRD - 1) ? (GRD - 1) : i1;
    w  = pos - (float)i0;
    o0 = i0 * (TRSTRIDE / 4);                     // offsets in f4 units (13 f4 per row)
    o1 = i1 * (TRSTRIDE / 4);
}

extern "C" __global__ void __launch_bounds__(THREADS, 1)
cp_interp_kernel(const float* __restrict__ xyz,
                 const float* __restrict__ p0,
                 const float* __restrict__ p1,
                 const float* __restrict__ p2,
                 float* __restrict__ out)
{
    extern __shared__ float smem[];
    const int tid = threadIdx.x;
    const float* gp[3] = { p0, p1, p2 };

    // ---- Stage each 48x300 table via async global->LDS, then transpose ----
    for (int a = 0; a < 3; ++a) {
        const float* src = gp[a];
        {
            unsigned stage_base = (unsigned)(uintptr_t)(&smem[0]);
            for (int k = tid; k < (STAGE_F / 4); k += THREADS) {
                unsigned lbyte = stage_base + (unsigned)k * 16u;   // LDS byte addr (per lane)
                unsigned gbyte = (unsigned)k * 16u;                // byte offset vs SGPR base
                asm volatile("global_load_async_to_lds_b128 %0, %1, %2"
                             :: "v"(lbyte), "v"(gbyte), "s"(src)
                             : "memory");
            }
#if __has_builtin(__builtin_amdgcn_s_wait_asynccnt)
            __builtin_amdgcn_s_wait_asynccnt(0);
#else
            asm volatile("s_wait_asynccnt 0" ::: "memory");
#endif
        }
        __syncthreads();
        // transpose staging [C][G] -> tr [G][TRSTRIDE]  (pad kills bank clustering)
        float* tr = &smem[STAGE_F + a * TR_F];
        for (int e = tid; e < STAGE_F; e += THREADS) {
            int c = e / GRD;
            int i = e - c * GRD;
            tr[i * TRSTRIDE + c] = smem[e];
        }
        __syncthreads();
    }

    const f4* T0 = (const f4*)&smem[STAGE_F + 0 * TR_F];
    const f4* T1 = (const f4*)&smem[STAGE_F + 1 * TR_F];
    const f4* T2 = (const f4*)&smem[STAGE_F + 2 * TR_F];

    // ---- 4 consecutive points per thread ----
    const int NG = NPTS / 4;                    // 500000, exact
    const int g  = blockIdx.x * THREADS + tid;
    if (g >= NG) return;
    const int n0 = g * 4;

    const f4* xq = (const f4*)(xyz + (size_t)n0 * 3);   // 16B-aligned (n0 % 4 == 0)
    f4 q0 = __builtin_nontemporal_load(xq + 0);         // x0 y0 z0 x1
    f4 q1 = __builtin_nontemporal_load(xq + 1);         // y1 z1 x2 y2
    f4 q2 = __builtin_nontemporal_load(xq + 2);         // z2 x3 y3 z3

    const float X[4] = { q0.x, q0.w, q1.z, q2.y };
    const float Y[4] = { q0.y, q1.x, q1.w, q2.z };
    const float Z[4] = { q0.z, q1.y, q2.x, q2.w };

    int   a0o[4], a1o[4], b0o[4], b1o[4], c0o[4], c1o[4];
    float wa[4], wb[4], wc[4];
#pragma unroll
    for (int p = 0; p < 4; ++p) {
        axis_prep(X[p], a0o[p], a1o[p], wa[p]);
        axis_prep(Y[p], b0o[p], b1o[p], wb[p]);
        axis_prep(Z[p], c0o[p], c1o[p], wc[p]);
    }

#pragma unroll 2
    for (int cc = 0; cc < NCOMP / 4; ++cc) {    // 4 components per iteration
        f4 r[4];
#pragma unroll
        for (int p = 0; p < 4; ++p) {
            f4 a0 = T0[a0o[p] + cc], a1 = T0[a1o[p] + cc];
            f4 b0 = T1[b0o[p] + cc], b1 = T1[b1o[p] + cc];
            f4 c0 = T2[c0o[p] + cc], c1 = T2[c1o[p] + cc];
            f4 va = a0 + wa[p] * (a1 - a0);
            f4 vb = b0 + wb[p] * (b1 - b0);
            f4 vc = c0 + wc[p] * (c1 - c0);
            r[p] = va * vb * vc;
        }
        // register transpose: one b128 NT store per component row
#pragma unroll
        for (int j = 0; j < 4; ++j) {
            f4 s = { r[0][j], r[1][j], r[2][j], r[3][j] };
            __builtin_nontemporal_store(
                s, (f4*)(out + (size_t)(cc * 4 + j) * NPTS + n0));
        }
    }
}

extern "C" void kernel_launch(void* const* d_in, const int* in_sizes, int n_in,
                              void* d_out, int out_size, void* d_ws, size_t ws_size,
                              hipStream_t stream) {
    (void)in_sizes; (void)n_in; (void)d_ws; (void)ws_size; (void)out_size;
    const float* xyz = (const float*)d_in[0];
    const float* p0  = (const float*)d_in[1];
    const float* p1  = (const float*)d_in[2];
    const float* p2  = (const float*)d_in[3];
    float* out = (float*)d_out;

    const int NG     = NPTS / 4;                           // 500000 point-quads
    const int blocks = (NG + THREADS - 1) / THREADS;       // 977
    const size_t shmem = (size_t)LDS_F * sizeof(float);    // 244800 B

    cp_interp_kernel<<<dim3(blocks), dim3(THREADS), shmem, stream>>>(
        xyz, p0, p1, p2, out);
}